// GCN_300647711437
// MI455X (gfx1250) — compile-verified
//
#include <hip/hip_runtime.h>

// GCN (3-layer, PyG gcn_norm) for MI455X / gfx1250.
// GEMMs run on the WMMA pipe with V_WMMA_F32_16X16X4_F32 (exact f32 math).
// - Layer-1 input zero-padded K:165->168 => unconditional aligned loads.
// - Software-pipelined K loop (double-buffered fragments) so WMMAs overlap
//   the next iteration's loads.
// - ReLU fused into consumer loads; layer-1 self-loop/bias seed fused into
//   the GEMM store epilogue.
// Aggregation is memory-bound scatter/gather via global_atomic_add_f32.

#define NNODES 200000
#define NEDGES 500000
#define FIN    165
#define FINP   168   /* FIN padded to multiple of 8 */
#define FHID   384
#define FOUT   2

typedef __attribute__((ext_vector_type(2))) float v2f;
typedef __attribute__((ext_vector_type(8))) float v8f;

// ---------------------------------------------------------------------------
// WMMA f32 GEMM: C[M,N] = act(A)[M,K] @ B[K,N]   (row-major, lda=K, ldb=ldc=N)
//   RELU:    apply max(x,0) to A elements as they are loaded.
//   FUSEAGG: additionally write AGG[m,n] = C[m,n]*dinv[m]^2 + bias[n].
// Requirements: K % 8 == 0 (K >= 8), N % 64 == 0.
// Block: 256 threads = 8 waves. Wave w: rows [m0, m0+32), cols [nblk, nblk+64)
// => block tile 256 x 64. EXEC is all-ones through every WMMA (M-edge handled
// by clamped loads + guarded stores, never by divergent early-exit).
// ---------------------------------------------------------------------------
template <bool RELU, bool FUSEAGG>
__global__ __launch_bounds__(256) void gemm_wmma_f32(
    const float* __restrict__ A, const float* __restrict__ B,
    const float* __restrict__ dinv, const float* __restrict__ bias,
    float* __restrict__ C, float* __restrict__ AGG, int M, int K, int N) {
  const int wave = threadIdx.x >> 5;
  const int lane = threadIdx.x & 31;
  const int half = lane >> 4;   // 0: K pair {0,1}/D rows r ; 1: K pair {2,3}/D rows r+8
  const int l16  = lane & 15;

  const int m0   = blockIdx.x * 256 + wave * 32;
  const int nblk = blockIdx.y * 64;

  int mA0 = m0 + l16;       if (mA0 >= M) mA0 = M - 1;   // clamp, masked at store
  int mA1 = m0 + 16 + l16;  if (mA1 >= M) mA1 = M - 1;
  const float* __restrict__ Ar0 = A + (size_t)mA0 * K + 2 * half;
  const float* __restrict__ Ar1 = A + (size_t)mA1 * K + 2 * half;
  const float* __restrict__ Bp  = B + (size_t)(2 * half) * N + nblk + l16;

  v8f a00 = {}, a01 = {}, a02 = {}, a03 = {};
  v8f a10 = {}, a11 = {}, a12 = {}, a13 = {};

  // Fragment loader (A: aligned b64; B: 8x b32, clause-friendly)
  auto loadFrag = [&](int kk, v2f& fA0, v2f& fA1,
                      v2f& fB0, v2f& fB1, v2f& fB2, v2f& fB3) {
    fA0 = *(const v2f*)(Ar0 + kk);
    fA1 = *(const v2f*)(Ar1 + kk);
    if (RELU) {
      fA0.x = fmaxf(fA0.x, 0.0f); fA0.y = fmaxf(fA0.y, 0.0f);
      fA1.x = fmaxf(fA1.x, 0.0f); fA1.y = fmaxf(fA1.y, 0.0f);
    }
    const float* __restrict__ p0 = Bp + (size_t)kk * N;
    const float* __restrict__ p1 = p0 + N;
    fB0.x = p0[0];  fB0.y = p1[0];
    fB1.x = p0[16]; fB1.y = p1[16];
    fB2.x = p0[32]; fB2.y = p1[32];
    fB3.x = p0[48]; fB3.y = p1[48];
  };

  auto mma8 = [&](const v2f& fA0, const v2f& fA1,
                  const v2f& fB0, const v2f& fB1, const v2f& fB2, const v2f& fB3) {
    a00 = __builtin_amdgcn_wmma_f32_16x16x4_f32(false, fA0, false, fB0, (short)0, a00, false, false);
    a01 = __builtin_amdgcn_wmma_f32_16x16x4_f32(false, fA0, false, fB1, (short)0, a01, false, false);
    a02 = __builtin_amdgcn_wmma_f32_16x16x4_f32(false, fA0, false, fB2, (short)0, a02, false, false);
    a03 = __builtin_amdgcn_wmma_f32_16x16x4_f32(false, fA0, false, fB3, (short)0, a03, false, false);
    a10 = __builtin_amdgcn_wmma_f32_16x16x4_f32(false, fA1, false, fB0, (short)0, a10, false, false);
    a11 = __builtin_amdgcn_wmma_f32_16x16x4_f32(false, fA1, false, fB1, (short)0, a11, false, false);
    a12 = __builtin_amdgcn_wmma_f32_16x16x4_f32(false, fA1, false, fB2, (short)0, a12, false, false);
    a13 = __builtin_amdgcn_wmma_f32_16x16x4_f32(false, fA1, false, fB3, (short)0, a13, false, false);
  };

  // Software-pipelined K loop: load(k+4) overlaps wmma(k).
  v2f cA0, cA1, cB0, cB1, cB2, cB3;
  loadFrag(0, cA0, cA1, cB0, cB1, cB2, cB3);
  int k0 = 0;
  for (; k0 + 4 < K; k0 += 4) {
    v2f nA0, nA1, nB0, nB1, nB2, nB3;
    loadFrag(k0 + 4, nA0, nA1, nB0, nB1, nB2, nB3);
    mma8(cA0, cA1, cB0, cB1, cB2, cB3);
    cA0 = nA0; cA1 = nA1;
    cB0 = nB0; cB1 = nB1; cB2 = nB2; cB3 = nB3;
  }
  mma8(cA0, cA1, cB0, cB1, cB2, cB3);

  // Epilogue. D layout: VGPR r -> row m0 + r + 8*half (+16 second fragment),
  // col nblk + j*16 + l16.
  float bv0 = 0.f, bv1 = 0.f, bv2 = 0.f, bv3 = 0.f;
  if (FUSEAGG) {
    bv0 = bias[nblk + l16];
    bv1 = bias[nblk + 16 + l16];
    bv2 = bias[nblk + 32 + l16];
    bv3 = bias[nblk + 48 + l16];
  }
#pragma unroll
  for (int r = 0; r < 8; ++r) {
    const int m = m0 + r + 8 * half;
    if (m < M) {
      const size_t off = (size_t)m * N + nblk + l16;
      float* cp = C + off;
      cp[0]  = a00[r];
      cp[16] = a01[r];
      cp[32] = a02[r];
      cp[48] = a03[r];
      if (FUSEAGG) {
        float d2 = dinv[m]; d2 *= d2;
        float* ap = AGG + off;
        ap[0]  = fmaf(a00[r], d2, bv0);
        ap[16] = fmaf(a01[r], d2, bv1);
        ap[32] = fmaf(a02[r], d2, bv2);
        ap[48] = fmaf(a03[r], d2, bv3);
      }
    }
    const int m2 = m + 16;
    if (m2 < M) {
      const size_t off = (size_t)m2 * N + nblk + l16;
      float* cp = C + off;
      cp[0]  = a10[r];
      cp[16] = a11[r];
      cp[32] = a12[r];
      cp[48] = a13[r];
      if (FUSEAGG) {
        float d2 = dinv[m2]; d2 *= d2;
        float* ap = AGG + off;
        ap[0]  = fmaf(a10[r], d2, bv0);
        ap[16] = fmaf(a11[r], d2, bv1);
        ap[32] = fmaf(a12[r], d2, bv2);
        ap[48] = fmaf(a13[r], d2, bv3);
      }
    }
  }
}

// Zero-pad x [N,165] -> xp [N,168]
__global__ void k_pad(const float* __restrict__ x, float* __restrict__ xp, unsigned n) {
  unsigned i = blockIdx.x * blockDim.x + threadIdx.x;
  unsigned total = n * FINP;
  if (i >= total) return;
  unsigned node = i / FINP;
  unsigned f    = i - node * FINP;
  xp[i] = (f < FIN) ? x[node * FIN + f] : 0.0f;
}

// ---------------------------------------------------------------------------
// gcn_norm pieces
// ---------------------------------------------------------------------------
__global__ void k_deg_init(float* __restrict__ deg, int n) {
  int i = blockIdx.x * blockDim.x + threadIdx.x;
  if (i < n) deg[i] = 1.0f;                      // self-loop contribution
}

__global__ void k_deg_accum(const int* __restrict__ col, float* __restrict__ deg, int E) {
  int e = blockIdx.x * blockDim.x + threadIdx.x;
  if (e < E) atomicAdd(&deg[col[e]], 1.0f);
}

__global__ void k_dinv(const float* __restrict__ deg, float* __restrict__ dinv, int n) {
  int i = blockIdx.x * blockDim.x + threadIdx.x;
  if (i < n) dinv[i] = rsqrtf(deg[i]);           // deg >= 1 always (self loops)
}

// agg[i,f] = h[i,f] * dinv[i]^2 + bias[f]  (used for layer 2, where the GEMM
// cannot fuse the agg store: its A operand aliases the agg buffer)
__global__ void k_init_agg(const float* __restrict__ h, const float* __restrict__ dinv,
                           const float* __restrict__ bias, float* __restrict__ agg,
                           unsigned n, unsigned F) {
  unsigned i = blockIdx.x * blockDim.x + threadIdx.x;
  unsigned total = n * F;
  if (i >= total) return;
  unsigned node = i / F;
  unsigned f    = i - node * F;
  float d = dinv[node];
  agg[i] = fmaf(h[i], d * d, bias[f]);
}

// One wave per edge: agg[col] += h[row] * (dinv[row]*dinv[col]), F = 384
__global__ __launch_bounds__(256) void k_scatter384(
    const float* __restrict__ h, const int* __restrict__ row, const int* __restrict__ col,
    const float* __restrict__ dinv, float* __restrict__ agg, int E) {
  int gw   = (blockIdx.x * blockDim.x + threadIdx.x) >> 5;
  int lane = threadIdx.x & 31;
  if (gw >= E) return;
  int r = row[gw], c = col[gw];
  float nrm = dinv[r] * dinv[c];
  const float4* __restrict__ hs = (const float4*)(h + (size_t)r * FHID);
  float* __restrict__ ad = agg + (size_t)c * FHID;
#pragma unroll
  for (int i = 0; i < 3; ++i) {
    float4 v = hs[lane + i * 32];
    int base = (lane + i * 32) * 4;
    atomicAdd(ad + base + 0, v.x * nrm);
    atomicAdd(ad + base + 1, v.y * nrm);
    atomicAdd(ad + base + 2, v.z * nrm);
    atomicAdd(ad + base + 3, v.w * nrm);
  }
}

// Layer-3 GEMM (N=2): one wave per node row, shuffle reduction, ReLU fused.
__global__ __launch_bounds__(256) void k_gemm_out2(
    const float* __restrict__ H, const float* __restrict__ W, float* __restrict__ T, int M) {
  int gw   = (blockIdx.x * blockDim.x + threadIdx.x) >> 5;
  int lane = threadIdx.x & 31;
  if (gw >= M) return;
  const float* __restrict__ hr = H + (size_t)gw * FHID;
  float a0 = 0.0f, a1 = 0.0f;
#pragma unroll
  for (int k = lane; k < FHID; k += 32) {
    float hv = fmaxf(hr[k], 0.0f);               // fused ReLU
    a0 = fmaf(hv, W[k * 2 + 0], a0);
    a1 = fmaf(hv, W[k * 2 + 1], a1);
  }
#pragma unroll
  for (int off = 16; off > 0; off >>= 1) {
    a0 += __shfl_down(a0, off, 32);
    a1 += __shfl_down(a1, off, 32);
  }
  if (lane == 0) { T[gw * 2 + 0] = a0; T[gw * 2 + 1] = a1; }
}

__global__ void k_init_agg2(const float* __restrict__ t, const float* __restrict__ dinv,
                            const float* __restrict__ bias, float* __restrict__ out, int n) {
  int i = blockIdx.x * blockDim.x + threadIdx.x;
  if (i >= n) return;
  float d2 = dinv[i] * dinv[i];
  out[i * 2 + 0] = fmaf(t[i * 2 + 0], d2, bias[0]);
  out[i * 2 + 1] = fmaf(t[i * 2 + 1], d2, bias[1]);
}

__global__ void k_scatter2(const float* __restrict__ t, const int* __restrict__ row,
                           const int* __restrict__ col, const float* __restrict__ dinv,
                           float* __restrict__ out, int E) {
  int e = blockIdx.x * blockDim.x + threadIdx.x;
  if (e >= E) return;
  int r = row[e], c = col[e];
  float nrm = dinv[r] * dinv[c];
  atomicAdd(out + c * 2 + 0, t[r * 2 + 0] * nrm);
  atomicAdd(out + c * 2 + 1, t[r * 2 + 1] * nrm);
}

// ---------------------------------------------------------------------------
extern "C" void kernel_launch(void* const* d_in, const int* in_sizes, int n_in,
                              void* d_out, int out_size, void* d_ws, size_t ws_size,
                              hipStream_t stream) {
  const float* x  = (const float*)d_in[0];
  const int*   ei = (const int*)d_in[1];        // [2, E] int32
  const float* W1 = (const float*)d_in[2];
  const float* b1 = (const float*)d_in[3];
  const float* W2 = (const float*)d_in[4];
  const float* b2 = (const float*)d_in[5];
  const float* W3 = (const float*)d_in[6];
  const float* b3 = (const float*)d_in[7];
  float* out = (float*)d_out;

  const int* erow = ei;                         // edge_index[0]
  const int* ecol = ei + NEDGES;                // edge_index[1]

  // Workspace carve-up (floats):
  // deg | dinv | xPad (N*168) | bufH (N*384) | bufA (N*384) | buf3 (N*2)
  float* deg  = (float*)d_ws;
  float* dinv = deg  + NNODES;
  float* xPad = dinv + NNODES;
  float* bufH = xPad + (size_t)NNODES * FINP;
  float* bufA = bufH + (size_t)NNODES * FHID;
  float* buf3 = bufA + (size_t)NNODES * FHID;

  const int T = 256;
  const int nodeBlocks  = (NNODES + T - 1) / T;
  const int edgeBlocks  = (NEDGES + T - 1) / T;
  const unsigned totalH = (unsigned)NNODES * FHID;
  const unsigned totalP = (unsigned)NNODES * FINP;
  const int elemBlocks  = (int)((totalH + T - 1) / T);
  const int padBlocks   = (int)((totalP + T - 1) / T);
  const int waveEdgeBlk = (NEDGES * 32 + T - 1) / T;   // wave-per-edge
  const int waveNodeBlk = (NNODES * 32 + T - 1) / T;   // wave-per-node
  const dim3 gemmGrid((NNODES + 255) / 256, FHID / 64);

  // gcn_norm: deg = 1 + segsum(ones, col); dinv = rsqrt(deg)
  k_deg_init<<<nodeBlocks, T, 0, stream>>>(deg, NNODES);
  k_deg_accum<<<edgeBlocks, T, 0, stream>>>(ecol, deg, NEDGES);
  k_dinv<<<nodeBlocks, T, 0, stream>>>(deg, dinv, NNODES);

  // Pad x to K=168 so GEMM1 hot loop is unconditional + aligned
  k_pad<<<padBlocks, T, 0, stream>>>(x, xPad, NNODES);

  // Layer 1: bufH = xPad @ W1 (agg seed fused into epilogue) ; scatter
  gemm_wmma_f32<false, true><<<gemmGrid, T, 0, stream>>>(
      xPad, W1, dinv, b1, bufH, bufA, NNODES, FINP, FHID);
  k_scatter384<<<waveEdgeBlk, T, 0, stream>>>(bufH, erow, ecol, dinv, bufA, NEDGES);

  // Layer 2: bufH = relu(bufA) @ W2 (ReLU fused into A loads) ; seed ; scatter
  gemm_wmma_f32<true, false><<<gemmGrid, T, 0, stream>>>(
      bufA, W2, dinv, b2, bufH, bufA /*unused*/, NNODES, FHID, FHID);
  k_init_agg<<<elemBlocks, T, 0, stream>>>(bufH, dinv, b2, bufA, NNODES, FHID);
  k_scatter384<<<waveEdgeBlk, T, 0, stream>>>(bufH, erow, ecol, dinv, bufA, NEDGES);

  // Layer 3: buf3 = relu(bufA) @ W3 ; out = scatter(buf3) + b3
  k_gemm_out2<<<waveNodeBlk, T, 0, stream>>>(bufA, W3, buf3, NNODES);
  k_init_agg2<<<nodeBlocks, T, 0, stream>>>(buf3, dinv, b3, out, NNODES);
  k_scatter2<<<edgeBlocks, T, 0, stream>>>(buf3, erow, ecol, dinv, out, NEDGES);
}